// MCPBRNN_PETconstraint_constant_OutLoss_53695681134739
// MI455X (gfx1250) — compile-verified
//
#include <hip/hip_runtime.h>
#include <math.h>

// Problem constants from the reference
#define BATCH 100000
#define SPIN  365
#define TRAIN 50000

// d_ws layout (floats): [0] = obsstd, [1] = oo, [2] = ol
//
// d_out layout (floats), flat concat of the reference's return tuple:
//   [ 0*B, 1*B) h_n        [ 1*B, 2*B) c_n       [ 2*B, 3*B) l_n
//   [ 3*B, 4*B) lc_n       [ 4*B, 5*B) bp_n(=0)  [ 5*B, 6*B) Gate_ib(=0)
//   [ 6*B, 7*B) Gate_oo    [ 7*B, 8*B) Gate_ol   [ 8*B, 9*B) Gate_olc
//   [ 9*B,10*B) Gate_f     [10*B,12*B) h_nout[B,2]  [12*B,13*B) obs_std

// ---------------------------------------------------------------------------
// Kernel 1: unbiased std over y_obs[SPIN:TRAIN] (double accumulation to avoid
// fp32 sum^2 cancellation) + softmax gates from the three scalar weights.
// ---------------------------------------------------------------------------
__global__ __launch_bounds__(256) void prep_kernel(
    const float* __restrict__ y_obs,
    const float* __restrict__ wom,
    const float* __restrict__ wlm,
    const float* __restrict__ wfm,
    float* __restrict__ ws)
{
    __shared__ double s_sum[256];
    __shared__ double s_sq[256];
    const int tid = threadIdx.x;
    double sum = 0.0, sq = 0.0;
    for (int i = SPIN + tid; i < TRAIN; i += 256) {
        double v = (double)y_obs[i];
        sum += v;
        sq  += v * v;
    }
    s_sum[tid] = sum;
    s_sq[tid]  = sq;
    __syncthreads();
    for (int s = 128; s > 0; s >>= 1) {
        if (tid < s) {
            s_sum[tid] += s_sum[tid + s];
            s_sq[tid]  += s_sq[tid + s];
        }
        __syncthreads();
    }
    if (tid == 0) {
        const double N    = (double)(TRAIN - SPIN);
        const double mean = s_sum[0] / N;
        const double var  = (s_sq[0] - s_sum[0] * mean) / (N - 1.0);  // ddof=1
        ws[0] = (float)sqrt(var);
        // gates: softmax over the three scalar weights
        float e0 = expf(wom[0]), e1 = expf(wlm[0]), e2 = expf(wfm[0]);
        float d  = e0 + e1 + e2;
        ws[1] = e0 / d;   // oo
        ws[2] = e1 / d;   // ol
    }
}

// ---------------------------------------------------------------------------
// Kernel 2: the inherently serial scan. One wave32: 32 lanes stage the next
// 32 (u1,u2) pairs into LDS and prefetch ahead; lane 0 runs the recurrence.
// The pre-update state c0[t] IS the c_n output column, so it streams straight
// into d_out — no scratch array needed.
// ---------------------------------------------------------------------------
__global__ __launch_bounds__(32) void scan_kernel(
    const float* __restrict__ x,          // [B, 1, 2]
    const int*   __restrict__ time_lag_p,
    const float* __restrict__ ws,
    float*       __restrict__ out)
{
    __shared__ float2 sx[32];
    const int   lane = threadIdx.x;
    const int   tl   = time_lag_p[0];
    const float oo   = ws[1];
    const float ol   = ws[2];
    const float one_m_oo = 1.0f - oo;
    const float2* __restrict__ x2 = (const float2*)x;
    float* __restrict__ c_col = out + BATCH;   // c_n column

    float c = 0.0f;
    for (int base = tl; base < BATCH; base += 32) {
        const int t = base + lane;
        if (t < BATCH) {
            sx[lane] = x2[t];                        // global_load_b64 -> ds_store
            if (t + 64 < BATCH)
                __builtin_prefetch(&x2[t + 64], 0, 0);  // global_prefetch_b8
        }
        __syncthreads();
        if (lane == 0) {
            const int n = (BATCH - base < 32) ? (BATCH - base) : 32;
            #pragma unroll 4
            for (int i = 0; i < n; ++i) {
                const float2 u = sx[i];
                float olc_raw;
                if (c > 0.0f) {
                    const float a = ol - u.y / c;
                    const float e = (a > 0.0f) ? a : (expf(a) - 1.0f);  // elu
                    olc_raw = ol - e;
                } else {
                    olc_raw = ol;
                }
                const float f = fmaxf(one_m_oo - olc_raw, 0.0f);
                c_col[base + i] = c;       // pre-update state == c_n[t]
                c = f * c + u.x;
            }
        }
        __syncthreads();
    }
}

// ---------------------------------------------------------------------------
// Kernel 3: fully parallel element-wise reconstruction of all other output
// columns from c0[t] (read back from the c_n column) and x[t].
// ---------------------------------------------------------------------------
__global__ __launch_bounds__(256) void outputs_kernel(
    const float* __restrict__ x,
    const int*   __restrict__ time_lag_p,
    const float* __restrict__ ws,
    float*       __restrict__ out)
{
    const int t = blockIdx.x * blockDim.x + threadIdx.x;
    if (t >= BATCH) return;
    const int   tl     = time_lag_p[0];
    const float obsstd = ws[0];
    const float oo     = ws[1];
    const float ol     = ws[2];

    float h = 0.0f, cc = 0.0f, l = 0.0f, lc = 0.0f;
    float g_olc = 0.0f, g_f = 0.0f, g_oo = 0.0f, g_ol = 0.0f, ostd = 0.0f;

    if (t >= tl) {
        const float c0 = out[BATCH + t];    // written by scan_kernel
        const float u2 = x[2 * t + 1];
        float olc_raw;
        if (c0 > 0.0f) {
            const float a = ol - u2 / c0;
            const float e = (a > 0.0f) ? a : (expf(a) - 1.0f);
            olc_raw = ol - e;
        } else {
            olc_raw = ol;
        }
        const float f   = fmaxf(1.0f - oo - olc_raw, 0.0f);
        const float olc = fmaxf(olc_raw, 0.0f);
        h = oo * c0;  cc = c0;  l = ol * c0;  lc = olc * c0;
        g_olc = olc;  g_f = f;  g_oo = oo;  g_ol = ol;  ostd = obsstd;
    }

    out[ 0 * BATCH + t] = h;       // h_n
    out[ 1 * BATCH + t] = cc;      // c_n  (rewrites same value / zero-pads)
    out[ 2 * BATCH + t] = l;       // l_n
    out[ 3 * BATCH + t] = lc;      // lc_n
    out[ 4 * BATCH + t] = 0.0f;    // bp_n
    out[ 5 * BATCH + t] = 0.0f;    // Gate_ib
    out[ 6 * BATCH + t] = g_oo;    // Gate_oo
    out[ 7 * BATCH + t] = g_ol;    // Gate_ol
    out[ 8 * BATCH + t] = g_olc;   // Gate_olc
    out[ 9 * BATCH + t] = g_f;     // Gate_f
    out[10 * BATCH + 2 * t]     = h;     // h_nout[:,0]
    out[10 * BATCH + 2 * t + 1] = ostd;  // h_nout[:,1]
    out[12 * BATCH + t] = ostd;    // obs_std
}

// ---------------------------------------------------------------------------
extern "C" void kernel_launch(void* const* d_in, const int* in_sizes, int n_in,
                              void* d_out, int out_size, void* d_ws, size_t ws_size,
                              hipStream_t stream) {
    (void)in_sizes; (void)n_in; (void)out_size; (void)ws_size;
    const float* x      = (const float*)d_in[0];  // [B,1,2]
    const float* y_obs  = (const float*)d_in[1];  // [B,1]
    const float* wom    = (const float*)d_in[2];  // [1,1]
    const float* wlm    = (const float*)d_in[3];  // [1,1]
    const float* wfm    = (const float*)d_in[4];  // [1,1]
    /* d_in[5] = epoch (unused) */
    const int*   tlag   = (const int*)d_in[6];    // scalar
    float* out = (float*)d_out;
    float* ws  = (float*)d_ws;

    prep_kernel<<<1, 256, 0, stream>>>(y_obs, wom, wlm, wfm, ws);
    scan_kernel<<<1, 32, 0, stream>>>(x, tlag, ws, out);
    outputs_kernel<<<(BATCH + 255) / 256, 256, 0, stream>>>(x, tlag, ws, out);
}